// TTTBlock_12764642803894
// MI455X (gfx1250) — compile-verified
//
#include <hip/hip_runtime.h>

typedef float v2f __attribute__((ext_vector_type(2)));
typedef float v4f __attribute__((ext_vector_type(4)));
typedef float v8f __attribute__((ext_vector_type(8)));

#define EPSF     1e-6f
#define BASE_LRF 0.01f

// B=4, H=16, T=512, K=16, D=64
#define Bc 4
#define Hc 16
#define Tc 512
#define Kc 16
#define Dc 64

// ---- gfx1250 async copy: global -> LDS, tracked with ASYNCcnt ----
__device__ __forceinline__ void async_ld_b128(void* lds, const void* gptr) {
    unsigned int       l = (unsigned int)(unsigned long long)(uintptr_t)lds; // LDS byte addr = low 32 bits
    unsigned long long g = (unsigned long long)(uintptr_t)gptr;
    asm volatile("global_load_async_to_lds_b128 %0, %1, off"
                 :: "v"(l), "v"(g) : "memory");
}
__device__ __forceinline__ void wait_async0() {
    asm volatile("s_wait_asynccnt 0" ::: "memory");
}

__device__ __forceinline__ v8f wmma_f32(v2f a, v2f b, v8f c) {
    return __builtin_amdgcn_wmma_f32_16x16x4_f32(false, a, false, b, (short)0, c, false, false);
}

// Parallel LayerNorm stats over 16 rows x 64 cols: 64 threads, 4 per row,
// combined with wave32 xor-shuffles (groups of 4 lanes stay inside a wave).
__device__ __forceinline__ void ln_stats_64(const float* sZp, float* sMu, float* sRs, int tid) {
    if (tid < 64) {
        int row = tid >> 2, q = tid & 3;
        const v4f* zr = (const v4f*)&sZp[row * Dc + q * 16];
        float s = 0.f, s2 = 0.f;
#pragma unroll
        for (int c = 0; c < 4; ++c) {
            v4f v = zr[c];
            s  += v.x + v.y + v.z + v.w;
            s2 += v.x * v.x + v.y * v.y + v.z * v.z + v.w * v.w;
        }
        s  += __shfl_xor(s, 1, 32);   s2 += __shfl_xor(s2, 1, 32);
        s  += __shfl_xor(s, 2, 32);   s2 += __shfl_xor(s2, 2, 32);
        if (q == 0) {
            float mu  = s * (1.0f / Dc);
            float var = s2 * (1.0f / Dc) - mu * mu;
            sMu[row] = mu;
            sRs[row] = rsqrtf(var + EPSF);
        }
    }
}

// One workgroup per (b,h) chain; 128 threads = 4 wave32 waves.
// Wave w owns feature columns [16w,16w+16). W1's 64x16 column slice lives in
// registers in WMMA-B layout (16 v2f chunks). Carried state: W1 regs + b1 LDS.
__launch_bounds__(128, 1)
__global__ void ttt_block_kernel(const float* __restrict__ XQ,
                                 const float* __restrict__ XK,
                                 const float* __restrict__ XV,
                                 const float* __restrict__ W1,
                                 const float* __restrict__ B1,
                                 const float* __restrict__ GAMMA,
                                 const float* __restrict__ BETA,
                                 float* __restrict__ OUT)
{
    __shared__ __align__(16) float sXQb[2][Kc * Dc];   // double-buffered async tiles
    __shared__ __align__(16) float sXKb[2][Kc * Dc];
    __shared__ __align__(16) float sXVb[2][Kc * Dc];
    __shared__ __align__(16) float sZ  [Kc * Dc];      // Z1 / Z1d staging
    __shared__ __align__(16) float sG  [Kc * Dc];      // grad_Z1
    __shared__ __align__(16) float sS  [Kc * Kc];      // S = xq @ xk^T (row i, col j)
    __shared__ float sB1 [Dc];
    __shared__ float sGB1[Dc];
    __shared__ float sGam[Dc];
    __shared__ float sBet[Dc];
    __shared__ float sMu [Kc];
    __shared__ float sRs [Kc];
    __shared__ float sEm, sEl;

    const int tid  = threadIdx.x;
    const int lane = tid & 31;
    const int wid  = tid >> 5;

    const int bh = blockIdx.x;          // 0..63
    const int b  = bh >> 4;
    const int h  = bh & 15;

    // WMMA wave32 operand helpers (16x16x4 f32)
    const int m16  = lane & 15;         // A-row / B-col / C-col index
    const int kb   = (lane >> 4) * 2;   // K sub-pair within chunk of 4
    const int crow = (lane >> 4) * 8;   // C/D row offset for upper half-wave
    const int col0 = wid * 16;          // this wave's column slice

    // ---- load W1 column slice into registers (B layout), init b1/gamma/beta ----
    v2f w[16];
    {
        const float* w1h = W1 + (size_t)h * Dc * Dc;
#pragma unroll
        for (int j = 0; j < 16; ++j) {
            int k0 = 4 * j + kb;
            w[j].x = w1h[k0 * Dc + col0 + m16];
            w[j].y = w1h[(k0 + 1) * Dc + col0 + m16];
        }
        if (tid < Dc) {
            sB1 [tid] = B1   [h * Dc + tid];
            sGam[tid] = GAMMA[h * Dc + tid];
            sBet[tid] = BETA [h * Dc + tid];
        }
    }

    const size_t baseBH = ((size_t)(b * Hc + h)) * Tc * Kc * Dc;

    // ---- prologue: async-stage t=0 tiles into buffer 0 ----
#pragma unroll
    for (int j = 0; j < 2; ++j) {
        int vi = tid + 128 * j;                     // 256 float4 per tile
        async_ld_b128(&sXQb[0][vi * 4], XQ + baseBH + vi * 4);
        async_ld_b128(&sXKb[0][vi * 4], XK + baseBH + vi * 4);
        async_ld_b128(&sXVb[0][vi * 4], XV + baseBH + vi * 4);
    }
    wait_async0();
    __syncthreads();

    for (int t = 0; t < Tc; ++t) {
        const size_t base = baseBH + (size_t)t * (Kc * Dc);
        const float* bQ = sXQb[t & 1];
        const float* bK = sXKb[t & 1];
        const float* bV = sXVb[t & 1];

        // ---- issue async loads for t+1 into the other buffer ----
        if (t + 1 < Tc) {
            const size_t nbase = base + Kc * Dc;
            float* nQ = sXQb[(t + 1) & 1];
            float* nK = sXKb[(t + 1) & 1];
            float* nV = sXVb[(t + 1) & 1];
#pragma unroll
            for (int j = 0; j < 2; ++j) {
                int vi = tid + 128 * j;
                async_ld_b128(&nQ[vi * 4], XQ + nbase + vi * 4);
                async_ld_b128(&nK[vi * 4], XK + nbase + vi * 4);
                async_ld_b128(&nV[vi * 4], XV + nbase + vi * 4);
            }
        }

        // ---- fused Z1 = xk@W1c and Zq = xq@W1c (shared B operands in regs) ----
        v8f zk = {}, zq = {};
#pragma unroll
        for (int j = 0; j < 16; ++j) {
            int k0 = 4 * j + kb;
            v2f ak = *(const v2f*)&bK[m16 * Dc + k0];
            v2f aq = *(const v2f*)&bQ[m16 * Dc + k0];
            zk = wmma_f32(ak, w[j], zk);
            zq = wmma_f32(aq, w[j], zq);
        }

        // ---- wave 3: P = xk @ xq^T (= S^T); A/B operands are the xk/xq pairs ----
        if (wid == 3) {
            v8f p = {};
#pragma unroll
            for (int j = 0; j < 16; ++j) {
                int k0 = 4 * j + kb;
                v2f ak = *(const v2f*)&bK[m16 * Dc + k0];
                v2f aq = *(const v2f*)&bQ[m16 * Dc + k0];
                p = wmma_f32(ak, aq, p);
            }
            // P[j][i] = S[i][j]; store row-major S: sS[i*16 + j]
#pragma unroll
            for (int r = 0; r < 8; ++r)
                sS[m16 * Kc + (r + crow)] = p[r];
        }

        // ---- wave 0: eta_k, em, el; 2 lanes per row + wave32 shuffles ----
        if (wid == 0) {
            int row = lane >> 1, half = lane & 1;
            const v4f* xk4 = (const v4f*)&bK[row * Dc + half * 32];
            float ss = 0.f;
#pragma unroll
            for (int q = 0; q < 8; ++q) {
                v4f v = xk4[q];
                ss += v.x * v.x + v.y * v.y + v.z * v.z + v.w * v.w;
            }
            ss += __shfl_xor(ss, 1, 32);          // combine halves of the row
            float nrm  = fmaxf(sqrtf(ss), EPSF);
            float etav = BASE_LRF / (1.0f + nrm); // duplicated per lane pair
            float s = etav;
            s += __shfl_xor(s, 1, 32);
            s += __shfl_xor(s, 2, 32);
            s += __shfl_xor(s, 4, 32);
            s += __shfl_xor(s, 8, 32);
            s += __shfl_xor(s, 16, 32);           // s = 2 * sum_k eta_k
            float elv = __shfl(etav, 30, 32);     // row 15 lives in lanes 30/31
            if (lane == 0) { sEm = s * (1.0f / 32.0f); sEl = elv; }
        }

        // ---- Z1 + b1 -> sZ ----
#pragma unroll
        for (int r = 0; r < 8; ++r) {
            int row = r + crow, col = col0 + m16;
            sZ[row * Dc + col] = zk[r] + sB1[col];
        }
        __syncthreads();                                      // (A)

        const float em = sEm;
        const float el = sEl;

        // ---- LN stats of Z1 ----
        ln_stats_64(sZ, sMu, sRs, tid);
        __syncthreads();                                      // (B)

        // ---- grad_Z1 = 2*(ln(Z1) - (xv - xk)) ----
#pragma unroll
        for (int j = 0; j < 2; ++j) {
            int vi = tid + 128 * j;
            int i0 = vi * 4, row = i0 >> 6, cb = i0 & 63;
            v4f z  = *(const v4f*)&sZ[i0];
            v4f xv = *(const v4f*)&bV[i0];
            v4f xk = *(const v4f*)&bK[i0];
            float mu = sMu[row], rs = sRs[row];
            v4f g;
#pragma unroll
            for (int c = 0; c < 4; ++c) {
                float lnv = sGam[cb + c] * (z[c] - mu) * rs + sBet[cb + c];
                g[c] = 2.0f * (lnv - (xv[c] - xk[c]));
            }
            *(v4f*)&sG[i0] = g;
        }
        __syncthreads();                                      // (C)

        // ---- grad_b1 (waves 0-1) ----
        if (tid < Dc) {
            float s = 0.f;
#pragma unroll
            for (int k = 0; k < Kc; ++k) s += sG[k * Dc + tid];
            sGB1[tid] = s;
        }

        // ---- SG = S @ grad_Z1 (16x16 tile per wave, K=16) ----
        v8f sg = {};
#pragma unroll
        for (int c = 0; c < 4; ++c) {
            int kk = 4 * c + kb;
            v2f a = *(const v2f*)&sS[m16 * Kc + kk];
            v2f bbv;
            bbv.x = sG[kk * Dc + col0 + m16];
            bbv.y = sG[(kk + 1) * Dc + col0 + m16];
            sg = wmma_f32(a, bbv, sg);
        }
        __syncthreads();                                      // (D)

        // ---- Z1d = Zq - em*SG + (b1 - em*grad_b1) -> sZ ----
#pragma unroll
        for (int r = 0; r < 8; ++r) {
            int row = r + crow, col = col0 + m16;
            sZ[row * Dc + col] = zq[r] - em * sg[r] + sB1[col] - em * sGB1[col];
        }
        __syncthreads();                                      // (E)

        // ---- LN stats of Z1d ----
        ln_stats_64(sZ, sMu, sRs, tid);
        __syncthreads();                                      // (F)

        // ---- out = xq + ln(Z1d) ----
        {
            v4f* go = (v4f*)(OUT + base);
#pragma unroll
            for (int j = 0; j < 2; ++j) {
                int vi = tid + 128 * j;
                int i0 = vi * 4, row = i0 >> 6, cb = i0 & 63;
                v4f z  = *(const v4f*)&sZ[i0];
                v4f xq = *(const v4f*)&bQ[i0];
                float mu = sMu[row], rs = sRs[row];
                v4f o;
#pragma unroll
                for (int c = 0; c < 4; ++c)
                    o[c] = xq[c] + sGam[cb + c] * (z[c] - mu) * rs + sBet[cb + c];
                go[vi] = o;
            }
        }

        // ---- carried state: rank-1 update directly on W1 registers ----
        {
            float g15 = sG[(Kc - 1) * Dc + col0 + m16];
            float eg  = el * g15;
#pragma unroll
            for (int j = 0; j < 16; ++j) {
                int k0 = 4 * j + kb;
                v2f xk15 = *(const v2f*)&bK[(Kc - 1) * Dc + k0];
                w[j].x -= eg * xk15.x;
                w[j].y -= eg * xk15.y;
            }
            if (tid < Dc) sB1[tid] -= el * sG[(Kc - 1) * Dc + tid];
        }

        // ---- tail: t+1 async tiles landed for my wave, then sync all ----
        wait_async0();
        __syncthreads();                                      // (G)
    }
}

extern "C" void kernel_launch(void* const* d_in, const int* in_sizes, int n_in,
                              void* d_out, int out_size, void* d_ws, size_t ws_size,
                              hipStream_t stream) {
    (void)in_sizes; (void)n_in; (void)out_size; (void)d_ws; (void)ws_size;
    const float* XQ    = (const float*)d_in[0];
    const float* XK    = (const float*)d_in[1];
    const float* XV    = (const float*)d_in[2];
    const float* W1    = (const float*)d_in[3];
    const float* B1    = (const float*)d_in[4];
    const float* GAMMA = (const float*)d_in[5];
    const float* BETA  = (const float*)d_in[6];
    float* OUT = (float*)d_out;

    dim3 grid(Bc * Hc);   // 64 independent (b,h) scan chains
    dim3 block(128);      // 4 wave32 waves
    ttt_block_kernel<<<grid, block, 0, stream>>>(XQ, XK, XV, W1, B1, GAMMA, BETA, OUT);
}